// QuantumNoiseFixed_26972394619665
// MI455X (gfx1250) — compile-verified
//
#include <hip/hip_runtime.h>
#include <math.h>

typedef __attribute__((ext_vector_type(2))) float v2f;
typedef __attribute__((ext_vector_type(8))) float v8f;

#define NQ 8
#define NL 3
#define DIM 256
#define WPB 8                    // waves per block (wave32)
#define THREADS (WPB * 32)

__device__ __forceinline__ float2 cmul(float2 a, float2 b) {
  return make_float2(a.x * b.x - a.y * b.y, a.x * b.y + a.y * b.x);
}

__global__ __launch_bounds__(THREADS) void qnoise_wmma_kernel(
    const float* __restrict__ weights,  // (NL, 2*NQ-1) = (3,15)
    const float* __restrict__ z1,       // (B,)
    const float* __restrict__ z2,       // (B,)
    float* __restrict__ out,            // (B, NQ)
    int batch) {
  // Batch-uniform tables, built once per block.
  __shared__ float  sU[NL * 2 * 256];        // [l][mat(0=Uh,1=Ul)][i*16+k]
  __shared__ float2 sD[NL * 256];            // diag phases exp(-i*ang/2)
  __shared__ float  sS[WPB][2][DIM];         // per-wave scratch: [re/im][16x16]

  const int t    = threadIdx.x;
  const int lane = t & 31;
  const int wave = t >> 5;
  const int m    = lane & 15;   // N / row-within-half index
  const int h    = lane >> 4;   // half-wave select

  // ---- setup: Kronecker-factored RY layer operators Uh, Ul ----
  for (int e = t; e < NL * 2 * 256; e += THREADS) {
    int l   = e >> 9;
    int mat = (e >> 8) & 1;     // 0: qubits 0-3 (rows), 1: qubits 4-7 (cols)
    int i   = (e >> 4) & 15;
    int k   = e & 15;
    float val = 1.0f;
#pragma unroll
    for (int j = 0; j < 4; ++j) {
      int q = mat * 4 + j;
      float th = 0.5f * weights[l * 15 + q];
      float c = cosf(th), s = sinf(th);
      int ib = (i >> (3 - j)) & 1;
      int kb = (k >> (3 - j)) & 1;
      val *= (ib == kb) ? c : ((ib == 0) ? -s : s);  // RY: [[c,-s],[s,c]]
    }
    sU[e] = val;
  }
  // ---- setup: fused RZZ diagonal per layer ----
  for (int e = t; e < NL * 256; e += THREADS) {
    int l = e >> 8;
    int idx = e & 255;
    float ang = 0.0f;
#pragma unroll
    for (int i2 = 0; i2 < 7; ++i2) {
      int b1 = (idx >> (7 - i2)) & 1;
      int b0 = (idx >> (6 - i2)) & 1;
      ang += weights[l * 15 + 8 + i2] * ((b1 == b0) ? 1.0f : -1.0f);
    }
    float ha = -0.5f * ang;                 // exp(-i*ang/2)
    sD[e] = make_float2(cosf(ha), sinf(ha));
  }
  __syncthreads();

  const int sample = blockIdx.x * WPB + wave;
  if (sample >= batch) return;              // wave-uniform; no later barriers

  // ---- initial product state: amp(idx) = prod_b v[bit_b] ----
  float a1 = asinf(z1[sample]);
  float a2 = asinf(z2[sample]);
  float ch = cosf(0.5f * a2), sh = sinf(0.5f * a2);
  float c1 = cosf(0.5f * a1), s1 = sinf(0.5f * a1);
  float2 v0 = make_float2(ch * c1, -sh * c1);   // e^{-i a2/2} cos(a1/2)
  float2 v1 = make_float2(ch * s1,  sh * s1);   // e^{+i a2/2} sin(a1/2)

  v8f Sre, Sim;
#pragma unroll
  for (int v = 0; v < 8; ++v) {
    int idx = (v + 8 * h) * 16 + m;
    float2 amp = make_float2(1.0f, 0.0f);
#pragma unroll
    for (int b = 0; b < 8; ++b)
      amp = cmul(amp, ((idx >> b) & 1) ? v1 : v0);
    Sre[v] = amp.x;
    Sim[v] = amp.y;
  }

  float* __restrict__ mre = sS[wave][0];
  float* __restrict__ mim = sS[wave][1];

  // ---- layers: S' = Uh * S * Ul^T, then diagonal phase ----
  for (int l = 0; l < NL; ++l) {
    const float* Uh = &sU[l * 512];
    const float* Ul = &sU[l * 512 + 256];

    // Preload real-operand fragments (A-frag of Uh; B-frag of Ul^T == A-pattern of Ul)
    v2f auh[4], bul[4];
#pragma unroll
    for (int kb = 0; kb < 4; ++kb) {
      auh[kb][0] = Uh[m * 16 + kb * 4 + 2 * h + 0];
      auh[kb][1] = Uh[m * 16 + kb * 4 + 2 * h + 1];
      bul[kb][0] = Ul[m * 16 + kb * 4 + 2 * h + 0];
      bul[kb][1] = Ul[m * 16 + kb * 4 + 2 * h + 1];
    }

    // Stage S in LDS (C/D layout -> row-major 16x16)
#pragma unroll
    for (int v = 0; v < 8; ++v) {
      mre[(v + 8 * h) * 16 + m] = Sre[v];
      mim[(v + 8 * h) * 16 + m] = Sim[v];
    }

    // Step 1: T = Uh * S  (real 16x16 times complex 16x16)
    v8f Tre = {0.f, 0.f, 0.f, 0.f, 0.f, 0.f, 0.f, 0.f};
    v8f Tim = {0.f, 0.f, 0.f, 0.f, 0.f, 0.f, 0.f, 0.f};
#pragma unroll
    for (int kb = 0; kb < 4; ++kb) {
      v2f bre, bim;
      bre[0] = mre[(kb * 4 + 0 + 2 * h) * 16 + m];
      bre[1] = mre[(kb * 4 + 1 + 2 * h) * 16 + m];
      bim[0] = mim[(kb * 4 + 0 + 2 * h) * 16 + m];
      bim[1] = mim[(kb * 4 + 1 + 2 * h) * 16 + m];
      Tre = __builtin_amdgcn_wmma_f32_16x16x4_f32(false, auh[kb], false, bre,
                                                  (short)0, Tre, false, false);
      Tim = __builtin_amdgcn_wmma_f32_16x16x4_f32(false, auh[kb], false, bim,
                                                  (short)0, Tim, false, false);
    }

    // Stage T in LDS for A-fragment reads
#pragma unroll
    for (int v = 0; v < 8; ++v) {
      mre[(v + 8 * h) * 16 + m] = Tre[v];
      mim[(v + 8 * h) * 16 + m] = Tim[v];
    }

    // Step 2: N = T * Ul^T
    v8f Nre = {0.f, 0.f, 0.f, 0.f, 0.f, 0.f, 0.f, 0.f};
    v8f Nim = {0.f, 0.f, 0.f, 0.f, 0.f, 0.f, 0.f, 0.f};
#pragma unroll
    for (int kb = 0; kb < 4; ++kb) {
      v2f are, aim;
      are[0] = mre[m * 16 + kb * 4 + 2 * h + 0];
      are[1] = mre[m * 16 + kb * 4 + 2 * h + 1];
      aim[0] = mim[m * 16 + kb * 4 + 2 * h + 0];
      aim[1] = mim[m * 16 + kb * 4 + 2 * h + 1];
      Nre = __builtin_amdgcn_wmma_f32_16x16x4_f32(false, are, false, bul[kb],
                                                  (short)0, Nre, false, false);
      Nim = __builtin_amdgcn_wmma_f32_16x16x4_f32(false, aim, false, bul[kb],
                                                  (short)0, Nim, false, false);
    }

    // Diagonal RZZ phase (elementwise complex multiply)
#pragma unroll
    for (int v = 0; v < 8; ++v) {
      int idx = (v + 8 * h) * 16 + m;
      float2 d = sD[l * 256 + idx];
      float re = Nre[v], im = Nim[v];
      Sre[v] = re * d.x - im * d.y;
      Sim[v] = re * d.y + im * d.x;
    }
  }

  // ---- probabilities -> <Z_q> projections ----
  float sq[NQ];
#pragma unroll
  for (int q = 0; q < NQ; ++q) sq[q] = 0.0f;
#pragma unroll
  for (int v = 0; v < 8; ++v) {
    int idx = (v + 8 * h) * 16 + m;
    float p = Sre[v] * Sre[v] + Sim[v] * Sim[v];
#pragma unroll
    for (int q = 0; q < NQ; ++q)
      sq[q] += ((idx >> (7 - q)) & 1) ? -p : p;
  }
  // wave32 butterfly reduction
#pragma unroll
  for (int off = 16; off > 0; off >>= 1) {
#pragma unroll
    for (int q = 0; q < NQ; ++q)
      sq[q] += __shfl_xor(sq[q], off, 32);
  }
  if (lane == 0) {
#pragma unroll
    for (int q = 0; q < NQ; ++q)
      out[sample * NQ + q] = sq[q];
  }
}

extern "C" void kernel_launch(void* const* d_in, const int* in_sizes, int n_in,
                              void* d_out, int out_size, void* d_ws, size_t ws_size,
                              hipStream_t stream) {
  const float* weights = (const float*)d_in[0];  // (3,15)
  const float* z1      = (const float*)d_in[1];  // (B,)
  const float* z2      = (const float*)d_in[2];  // (B,)
  float* out           = (float*)d_out;          // (B, 8)
  int batch = in_sizes[1];
  int blocks = (batch + WPB - 1) / WPB;
  qnoise_wmma_kernel<<<blocks, THREADS, 0, stream>>>(weights, z1, z2, out, batch);
}